// ICApproxLayer_1176821039626
// MI455X (gfx1250) — compile-verified
//
#include <hip/hip_runtime.h>
#include <stdint.h>

// ---------------------------------------------------------------------------
// Independent-Cascade approximation, 10 steps, on MI455X.
// Strategy:
//  * Build CSR grouped by dst once per call (hist + single-block scan +
//    permute). Working set ~31MB -> fully resident in the 192MB L2.
//  * Per step, ONE fused kernel: each 256-thread block owns 256 consecutive
//    nodes; its in-edges are a contiguous CSR range. Edge streams are staged
//    into LDS with gfx1250 async loads (ASYNCcnt double buffer), accumulated
//    with ds_add_f32 into a per-block s_delta[256], then the node update
//    (p_t = run*(1-exp(-delta)); run *= 1-p_t) runs in the same kernel.
//  * No global fp atomics at all in the hot loop; p is double-buffered.
// ---------------------------------------------------------------------------

#define STEP_BLOCK 256
#define GROUP      256                       // nodes per block
#define EPT        4                         // edges per thread per chunk
#define ECHUNK     (STEP_BLOCK * EPT)        // 1024 edges per block-iteration
#define SCAN_BLOCK 1024
#define NUM_STEPS  10

__global__ void k_zero(int* __restrict__ a, int N) {
  int i = blockIdx.x * blockDim.x + threadIdx.x;
  if (i < N) a[i] = 0;
}

__global__ void k_hist(const long long* __restrict__ ei, int* __restrict__ cnt,
                       int E) {
  int i = blockIdx.x * blockDim.x + threadIdx.x;
  if (i < E) atomicAdd(&cnt[(int)ei[(long long)E + i]], 1);
}

// Single-block exclusive scan: rowptr[0..N] and fill[i] = rowptr[i].
__global__ __launch_bounds__(SCAN_BLOCK)
void k_scan(const int* __restrict__ cnt, int* __restrict__ rowptr,
            int* __restrict__ fill, int N) {
  __shared__ int s_part[SCAN_BLOCK];
  const int tid = threadIdx.x;
  const int per = (N + SCAN_BLOCK - 1) / SCAN_BLOCK;
  const int beg = tid * per;
  const int end = min(beg + per, N);
  int sum = 0;
  for (int i = beg; i < end; ++i) sum += cnt[i];
  s_part[tid] = sum;
  __syncthreads();
  for (int offd = 1; offd < SCAN_BLOCK; offd <<= 1) {  // Hillis-Steele incl.
    int v = (tid >= offd) ? s_part[tid - offd] : 0;
    __syncthreads();
    s_part[tid] += v;
    __syncthreads();
  }
  int runv = (tid == 0) ? 0 : s_part[tid - 1];
  for (int i = beg; i < end; ++i) {
    rowptr[i] = runv;
    fill[i]   = runv;
    runv += cnt[i];
  }
  if (end == N && beg <= N) rowptr[N] = runv;  // total E
}

__global__ void k_fill(const long long* __restrict__ ei,
                       const float* __restrict__ w, int* __restrict__ fill,
                       int* __restrict__ csr_src, float* __restrict__ csr_prob,
                       unsigned char* __restrict__ csr_ldst, int E) {
  int i = blockIdx.x * blockDim.x + threadIdx.x;
  if (i < E) {
    int s = (int)ei[i];
    int d = (int)ei[(long long)E + i];
    int pos = atomicAdd(&fill[d], 1);
    csr_src[pos]  = s;
    csr_prob[pos] = w[i];
    csr_ldst[pos] = (unsigned char)(d & (GROUP - 1));  // local node index
  }
}

__global__ void k_init(const float* __restrict__ prior, float* __restrict__ p,
                       float* __restrict__ run, int N) {
  int i = blockIdx.x * blockDim.x + threadIdx.x;
  if (i < N) {
    float pr = prior[i];
    p[i]   = pr;
    run[i] = 1.0f - pr;
  }
}

// Fused per-step kernel: stream this block's CSR edge range via async-to-LDS,
// bin with ds_add_f32 into s_delta, then update the block's 256 nodes.
__global__ __launch_bounds__(STEP_BLOCK)
void k_step(const int* __restrict__ rowptr, const int* __restrict__ csr_src,
            const float* __restrict__ csr_prob,
            const unsigned char* __restrict__ csr_ldst,
            const float* __restrict__ pOld, float* __restrict__ run,
            float* __restrict__ pNew, float* __restrict__ out, int N) {
  __shared__ float    s_delta[GROUP];
  __shared__ int4     s_src[2][STEP_BLOCK];   // 8 KB
  __shared__ float4   s_prb[2][STEP_BLOCK];   // 8 KB
  __shared__ unsigned s_li [2][STEP_BLOCK];   // 2 KB (4 x u8 per lane)

  const int tid   = threadIdx.x;
  const int nbase = blockIdx.x * GROUP;
  s_delta[tid] = 0.0f;

  const int eBeg = rowptr[nbase];
  const int eEnd = rowptr[min(nbase + GROUP, N)];
  const int eA   = eBeg & ~3;                     // 16B-align the stream
  const int nChunks = (eEnd - eA + ECHUNK - 1) / ECHUNK;

  const uint64_t sbase = (uint64_t)csr_src;
  const uint64_t pbase = (uint64_t)csr_prob;
  const uint64_t lbase = (uint64_t)csr_ldst;

  auto issue = [&](int buf, int chunk) {
    unsigned e    = (unsigned)(eA + chunk * ECHUNK + tid * EPT);  // edge index
    unsigned off4 = e * 4u;                                       // 4B arrays
    unsigned lsrc = (unsigned)(uintptr_t)&s_src[buf][tid];
    unsigned lprb = (unsigned)(uintptr_t)&s_prb[buf][tid];
    unsigned lli  = (unsigned)(uintptr_t)&s_li[buf][tid];
    asm volatile(
        "global_load_async_to_lds_b128 %0, %3, %5\n\t"
        "global_load_async_to_lds_b128 %1, %3, %6\n\t"
        "global_load_async_to_lds_b32  %2, %4, %7"
        :
        : "v"(lsrc), "v"(lprb), "v"(lli), "v"(off4), "v"(e),
          "s"(sbase), "s"(pbase), "s"(lbase)
        : "memory");
  };

  if (nChunks > 0) issue(0, 0);
  __syncthreads();  // s_delta zeroing visible before any ds_add

  for (int c = 0, buf = 0; c < nChunks; ++c, buf ^= 1) {
    if (c + 1 < nChunks) {
      issue(buf ^ 1, c + 1);
      // ASYNCcnt completes in order: <=3 outstanding => prev trio landed.
      asm volatile("s_wait_asynccnt 3" ::: "memory");
    } else {
      asm volatile("s_wait_asynccnt 0" ::: "memory");
    }
    const int e0 = eA + c * ECHUNK + tid * EPT;
    int4     s4 = s_src[buf][tid];
    float4   w4 = s_prb[buf][tid];
    unsigned l4 = s_li[buf][tid];
    if (e0 >= eBeg && e0 + EPT <= eEnd) {        // all 4 edges valid
      atomicAdd(&s_delta[(l4 >> 0)  & 255u], w4.x * pOld[s4.x]);
      atomicAdd(&s_delta[(l4 >> 8)  & 255u], w4.y * pOld[s4.y]);
      atomicAdd(&s_delta[(l4 >> 16) & 255u], w4.z * pOld[s4.z]);
      atomicAdd(&s_delta[(l4 >> 24) & 255u], w4.w * pOld[s4.w]);
    } else {                                      // range-guarded tail/head
      if (e0 + 0 >= eBeg && e0 + 0 < eEnd)
        atomicAdd(&s_delta[(l4 >> 0)  & 255u], w4.x * pOld[s4.x]);
      if (e0 + 1 >= eBeg && e0 + 1 < eEnd)
        atomicAdd(&s_delta[(l4 >> 8)  & 255u], w4.y * pOld[s4.y]);
      if (e0 + 2 >= eBeg && e0 + 2 < eEnd)
        atomicAdd(&s_delta[(l4 >> 16) & 255u], w4.z * pOld[s4.z]);
      if (e0 + 3 >= eBeg && e0 + 3 < eEnd)
        atomicAdd(&s_delta[(l4 >> 24) & 255u], w4.w * pOld[s4.w]);
    }
  }
  __syncthreads();  // all ds_add_f32 for this block complete

  const int i = nbase + tid;
  if (i < N) {
    float d  = s_delta[tid];
    float r  = run[i];
    float pt = r * (1.0f - __expf(-d));
    float rn = r * (1.0f - pt);
    run[i]  = rn;
    pNew[i] = pt;
    if (out) out[i] = 1.0f - rn;   // final step only
  }
}

extern "C" void kernel_launch(void* const* d_in, const int* in_sizes, int n_in,
                              void* d_out, int out_size, void* d_ws,
                              size_t ws_size, hipStream_t stream) {
  const float*     prior = (const float*)d_in[0];
  const long long* ei    = (const long long*)d_in[1];  // int64 [2, E]
  const float*     eprob = (const float*)d_in[2];
  const int N = in_sizes[0];
  const int E = in_sizes[2];

  // Workspace carve-out (~31 MB, 64B-aligned slices; CSR arrays padded by one
  // chunk so async prefetch reads never leave the workspace).
  char*  w   = (char*)d_ws;
  size_t off = 0;
  auto alloc = [&](size_t bytes) -> void* {
    void* p = w + off;
    off = (off + bytes + 63) & ~(size_t)63;
    return p;
  };
  int*           cnt      = (int*)alloc((size_t)N * 4);
  int*           fill     = (int*)alloc((size_t)N * 4);
  int*           rowptr   = (int*)alloc(((size_t)N + 1) * 4);
  int*           csr_src  = (int*)alloc(((size_t)E + ECHUNK) * 4);
  float*         csr_prob = (float*)alloc(((size_t)E + ECHUNK) * 4);
  unsigned char* csr_ldst = (unsigned char*)alloc((size_t)E + ECHUNK);
  float*         pA       = (float*)alloc((size_t)N * 4);
  float*         pB       = (float*)alloc((size_t)N * 4);
  float*         run      = (float*)alloc((size_t)N * 4);
  (void)ws_size;

  const int nbN = (N + 255) / 256;
  const int nbE = (E + 255) / 256;

  // One-time CSR build (re-run every call: deterministic work sequence).
  k_zero<<<nbN, 256, 0, stream>>>(cnt, N);
  k_hist<<<nbE, 256, 0, stream>>>(ei, cnt, E);
  k_scan<<<1, SCAN_BLOCK, 0, stream>>>(cnt, rowptr, fill, N);
  k_fill<<<nbE, 256, 0, stream>>>(ei, eprob, fill, csr_src, csr_prob,
                                  csr_ldst, E);
  k_init<<<nbN, 256, 0, stream>>>(prior, pA, run, N);

  const int nGroups = (N + GROUP - 1) / GROUP;
  float* pOld = pA;
  float* pNew = pB;
  for (int t = 0; t < NUM_STEPS; ++t) {
    float* o = (t == NUM_STEPS - 1) ? (float*)d_out : nullptr;
    k_step<<<nGroups, STEP_BLOCK, 0, stream>>>(rowptr, csr_src, csr_prob,
                                               csr_ldst, pOld, run, pNew, o, N);
    float* tmp = pOld; pOld = pNew; pNew = tmp;
  }
}